// BahdanauAttention_48395691491566
// MI455X (gfx1250) — compile-verified
//
#include <hip/hip_runtime.h>
#include <hip/hip_bf16.h>

#define B_ 8
#define T_ 256
#define S_ 256
#define H_ 512
#define TT 16   // t-rows per fused block (one WMMA M tile)

typedef __attribute__((ext_vector_type(2))) float v2f;
typedef __attribute__((ext_vector_type(8))) float v8f;

__device__ __forceinline__ v8f wmma_f32_4(v2f a, v2f b, v8f c) {
  // V_WMMA_F32_16X16X4_F32 : 8-arg form (neg_a, A, neg_b, B, c_mod, C, reuse_a, reuse_b)
  return __builtin_amdgcn_wmma_f32_16x16x4_f32(false, a, false, b, (short)0, c,
                                               false, false);
}

__device__ __forceinline__ float fast_tanh(float x) {
#if __has_builtin(__builtin_amdgcn_tanhf)
  return __builtin_amdgcn_tanhf(x);            // gfx1250 v_tanh_f32
#elif __has_builtin(__builtin_amdgcn_tanh_f32)
  return __builtin_amdgcn_tanh_f32(x);
#else
  float e = __expf(2.0f * x);                  // v_exp_f32 path
  return 1.0f - 2.0f / (e + 1.0f);
#endif
}

// ---------------------------------------------------------------------------
// Kernel 1: enc_fT[b][h][s] = sum_k W_h[h][k] * enc[b][s][k]   (stored (B,H,S))
// One 16x16 C tile per wave. K = 512 -> 128 wmma_f32_16x16x4 steps.
// ---------------------------------------------------------------------------
__global__ __launch_bounds__(256) void encfT_gemm(const float* __restrict__ enc,
                                                  const float* __restrict__ Wh,
                                                  float* __restrict__ encfT) {
  const int wave = blockIdx.x * 8 + (threadIdx.x >> 5);  // 4096 waves total
  const int lane = threadIdx.x & 31;
  const int ST = S_ / 16, HT = H_ / 16;
  int tmp = wave;
  const int st = tmp % ST; tmp /= ST;
  const int ht = tmp % HT; tmp /= HT;
  const int b = tmp;                                     // 0..7
  const int m0 = ht * 16, n0 = st * 16;
  const int l16 = lane & 15;
  const int ko  = (lane >> 4) << 1;                      // 0 or 2 (K half)

  const float* Arow = Wh + (size_t)(m0 + l16) * H_;                  // A[m][k]
  const float* Brow = enc + ((size_t)b * S_ + (n0 + l16)) * H_;      // B[k][n]=enc[b][n][k]

  v8f c = {};
  for (int k = 0; k < H_; k += 4) {
    v2f a, bb;
    a.x  = Arow[k + ko];  a.y  = Arow[k + ko + 1];
    bb.x = Brow[k + ko];  bb.y = Brow[k + ko + 1];
    c = wmma_f32_4(a, bb, c);
  }
  const int rbase = (lane < 16) ? 0 : 8;
  const int scol  = n0 + l16;
#pragma unroll
  for (int r = 0; r < 8; ++r) {
    int m = m0 + rbase + r;
    encfT[((size_t)b * H_ + m) * S_ + scol] = c[r];
  }
}

// ---------------------------------------------------------------------------
// Kernel 2: qry_f[b][t][h] = sum_k query[b][t][k] * W_s[h][k]   (stored (B,T,H))
// ---------------------------------------------------------------------------
__global__ __launch_bounds__(256) void qryf_gemm(const float* __restrict__ query,
                                                 const float* __restrict__ Ws,
                                                 float* __restrict__ qryf) {
  const int wave = blockIdx.x * 8 + (threadIdx.x >> 5);  // 4096 waves total
  const int lane = threadIdx.x & 31;
  const int NT = H_ / 16, MT = T_ / 16;
  int tmp = wave;
  const int nt = tmp % NT; tmp /= NT;
  const int mt = tmp % MT; tmp /= MT;
  const int b = tmp;
  const int m0 = mt * 16, n0 = nt * 16;
  const int l16 = lane & 15;
  const int ko  = (lane >> 4) << 1;

  const float* Arow = query + ((size_t)b * T_ + (m0 + l16)) * H_;    // A[m][k]
  const float* Brow = Ws + (size_t)(n0 + l16) * H_;                  // B[k][n]=Ws[n][k]

  v8f c = {};
  for (int k = 0; k < H_; k += 4) {
    v2f a, bb;
    a.x  = Arow[k + ko];  a.y  = Arow[k + ko + 1];
    bb.x = Brow[k + ko];  bb.y = Brow[k + ko + 1];
    c = wmma_f32_4(a, bb, c);
  }
  const int rbase = (lane < 16) ? 0 : 8;
  const int hcol  = n0 + l16;
#pragma unroll
  for (int r = 0; r < 8; ++r) {
    int t = m0 + rbase + r;
    qryf[((size_t)b * T_ + t) * H_ + hcol] = c[r];
  }
}

// ---------------------------------------------------------------------------
// Kernel 3 (fused): per (b, 16-t tile)
//   scores -> mask -> softmax -> attn_out via in-block WMMA
// Thread layout: 256 threads = 8 waves. Score stage maps thread -> s.
// ---------------------------------------------------------------------------
__global__ __launch_bounds__(256) void attn_fused(const float* __restrict__ encfT,
                                                  const float* __restrict__ qryf,
                                                  const float* __restrict__ enc,
                                                  const float* __restrict__ vvec,
                                                  const int* __restrict__ lens,
                                                  float* __restrict__ out) {
  __shared__ float q_lds[TT][H_];        // 32 KB, broadcast reads (conflict-free)
  __shared__ float w_lds[TT][S_ + 1];    // padded: WMMA-A reads stride 257 -> no bank conflicts

  const int tid = threadIdx.x;
  const int b   = blockIdx.x / (T_ / TT);
  const int t0  = (blockIdx.x % (T_ / TT)) * TT;

  // stage query-projection tile into LDS
  for (int i = tid; i < TT * H_; i += 256) {
    int tt = i >> 9, h = i & (H_ - 1);
    q_lds[tt][h] = qryf[((size_t)b * T_ + t0 + tt) * H_ + h];
  }
  __syncthreads();

  // ---- score stage: thread owns one s; 16 tanh per enc_fT load ----
  const int s = tid;
  float acc[TT];
#pragma unroll
  for (int tt = 0; tt < TT; ++tt) acc[tt] = 0.0f;

  const float* ecol = encfT + (size_t)b * H_ * S_ + s;   // coalesced across lanes
  for (int h = 0; h < H_; ++h) {
    float e  = ecol[(size_t)h * S_];
    float vh = vvec[h];                                  // uniform -> scalar load
#pragma unroll
    for (int tt = 0; tt < TT; ++tt)
      acc[tt] = fmaf(vh, fast_tanh(e + q_lds[tt][h]), acc[tt]);
  }

  const int  len   = lens[b];
  const bool valid = (s < len);
#pragma unroll
  for (int tt = 0; tt < TT; ++tt)
    w_lds[tt][s] = valid ? acc[tt] : -__builtin_inff();
  __syncthreads();

  // ---- masked softmax: 16 threads per t-row, shuffle reductions ----
  {
    const int g = tid >> 4;       // row 0..15
    const int l = tid & 15;
    float m = -__builtin_inff();
#pragma unroll
    for (int i = 0; i < 16; ++i) m = fmaxf(m, w_lds[g][l + 16 * i]);
#pragma unroll
    for (int off = 8; off >= 1; off >>= 1) m = fmaxf(m, __shfl_xor(m, off, 32));
    float ssum = 0.0f;
#pragma unroll
    for (int i = 0; i < 16; ++i) {
      float e = __expf(w_lds[g][l + 16 * i] - m);   // exp(-inf)=0 handles mask
      w_lds[g][l + 16 * i] = e;
      ssum += e;
    }
#pragma unroll
    for (int off = 8; off >= 1; off >>= 1) ssum += __shfl_xor(ssum, off, 32);
    const float inv = 1.0f / ssum;
#pragma unroll
    for (int i = 0; i < 16; ++i) w_lds[g][l + 16 * i] *= inv;
  }
  __syncthreads();

  // ---- output: C(16 x 512) = W(16 x 256) @ enc[b](256 x 512) via WMMA ----
  const int wave = tid >> 5;            // 0..7, each owns 64 output columns
  const int lane = tid & 31;
  const int l16  = lane & 15;
  const int ko   = (lane >> 4) << 1;
  const int n0b  = wave * 64;

  v8f cacc[4] = {};
  for (int k = 0; k < S_; k += 4) {
    v2f a;
    a.x = w_lds[l16][k + ko];
    a.y = w_lds[l16][k + ko + 1];
#pragma unroll
    for (int nt = 0; nt < 4; ++nt) {
      const int n = n0b + nt * 16 + l16;
      const float* bp = enc + ((size_t)b * S_ + k + ko) * H_ + n;
      v2f bb;
      bb.x = bp[0];
      bb.y = bp[H_];
      cacc[nt] = wmma_f32_4(a, bb, cacc[nt]);
    }
  }
  const int rbase = (lane < 16) ? 0 : 8;
#pragma unroll
  for (int nt = 0; nt < 4; ++nt) {
    const int n = n0b + nt * 16 + l16;
#pragma unroll
    for (int r = 0; r < 8; ++r) {
      const int t = t0 + rbase + r;
      out[((size_t)b * T_ + t) * H_ + n] = cacc[nt][r];
    }
  }
}

// ---------------------------------------------------------------------------
extern "C" void kernel_launch(void* const* d_in, const int* in_sizes, int n_in,
                              void* d_out, int out_size, void* d_ws, size_t ws_size,
                              hipStream_t stream) {
  const float* query = (const float*)d_in[0];   // (B,T,H)
  const float* enc   = (const float*)d_in[1];   // (B,S,H)
  const int*   lens  = (const int*)d_in[2];     // (B,)
  const float* Wh    = (const float*)d_in[3];   // (H,H)
  const float* Ws    = (const float*)d_in[4];   // (H,H)
  const float* vvec  = (const float*)d_in[5];   // (H,)
  float* out = (float*)d_out;                   // (B,T,H)

  float* encfT = (float*)d_ws;                         // B*H*S floats (4 MB)
  float* qryf  = encfT + (size_t)B_ * H_ * S_;         // B*T*H floats (4 MB)

  // 4096 wave-tiles each, 8 waves/block -> 512 blocks
  encfT_gemm<<<512, 256, 0, stream>>>(enc, Wh, encfT);
  qryf_gemm <<<512, 256, 0, stream>>>(query, Ws, qryf);
  attn_fused<<<B_ * (T_ / TT), 256, 0, stream>>>(encfT, qryf, enc, vvec, lens, out);
}